// OverFit_43198781063824
// MI455X (gfx1250) — compile-verified
//
#include <hip/hip_runtime.h>

// ---------------------------------------------------------------------------
// GCN forward, 5 layers, gfx1250 (MI455X).
//   out = D^-1/2 (A+I) D^-1/2 (X W) + b  per layer (ReLU on layers 1..4)
// Refactor:  hs = (X@W) * dinv[row]   (fused in WMMA epilogue)
//            agg = hs (self loop) + scatter_add(hs[src] -> dst)
//            out = agg * dinv[row] + b (+ReLU)
// GEMM: V_WMMA_F32_16X16X4_F32, branchless K-loop via zero-padded transposed
// weights (Wt[outPad][Kpad]) and padded-stride intermediate buffers.
// ---------------------------------------------------------------------------

typedef float v2f __attribute__((ext_vector_type(2)));
typedef float v8f __attribute__((ext_vector_type(8)));

#define N_NODES 100000
#define N_EDGES 1600000

// padded dims: outPad = ceil16(out); Kpad[L] = input stride of layer L
// dims: 512 -> 200 -> 175 -> 125 -> 75 -> 50
// outPad:     208    176    128    80     64
// Kpad :  512 208    176    128    80
#define MAX_OUTPAD 208

// ---------------- degree / dinv ----------------
__global__ void k_init_deg(float* __restrict__ deg) {
    int i = blockIdx.x * blockDim.x + threadIdx.x;
    if (i < N_NODES) deg[i] = 1.0f;   // self loop
}

__global__ void k_accum_deg(const int* __restrict__ dst, float* __restrict__ deg) {
    int i = blockIdx.x * blockDim.x + threadIdx.x;
    if (i < N_EDGES) atomicAdd(&deg[dst[i]], 1.0f);
}

__global__ void k_rsqrt_deg(float* __restrict__ deg) {
    int i = blockIdx.x * blockDim.x + threadIdx.x;
    if (i < N_NODES) deg[i] = rsqrtf(deg[i]);
}

// ---- zero-padded transposed weights: Wt[n*Kpad + k] = W[k*out + n] --------
__global__ void k_pad_weights(const float* __restrict__ W, float* __restrict__ Wt,
                              int K, int out, int Kpad, int outPad) {
    int t = blockIdx.x * blockDim.x + threadIdx.x;
    int total = Kpad * outPad;
    if (t >= total) return;
    int n = t / Kpad;
    int k = t - n * Kpad;
    Wt[t] = (k < K && n < out) ? W[(size_t)k * out + n] : 0.0f;
}

// ---------------- WMMA GEMM (branchless) -----------------------------------
// One wave -> one 16x16 C tile. 4 waves/block, consecutive waves take
// consecutive column tiles of one 16-row strip (A reuse through L0/L2).
// A (16x4 f32): lanes 0-15 rows M=0..15; VGPR0=K(+0/+2), VGPR1=K(+1/+3).
// B (4x16 f32): lanes 0-15 cols N=0..15; same K split per lane half.
// C (16x16 f32): lane = col; VGPR i = row i (+8 for lanes 16-31).
__global__ __launch_bounds__(128)
void k_gemm_scale_wmma(const float* __restrict__ X, const float* __restrict__ Wt,
                       const float* __restrict__ dinv,
                       float* __restrict__ hs, float* __restrict__ agg,
                       int Kpad, int outPad, int nTiles) {
    const int wave = threadIdx.x >> 5;
    const int lane = threadIdx.x & 31;
    const int tile = blockIdx.x * 4 + wave;
    const int mTiles = N_NODES / 16;              // 6250, exact
    if (tile >= mTiles * nTiles) return;

    const int mt = tile / nTiles;
    const int nt = tile - mt * nTiles;
    const int mBase = mt * 16;
    const int n = nt * 16 + (lane & 15);          // col of B/C for this lane
    const int lg = lane >> 4;                     // lane half -> K offset 0/2

    const float* __restrict__ arow = X  + (size_t)(mBase + (lane & 15)) * Kpad + 2 * lg;
    const float* __restrict__ brow = Wt + (size_t)n * Kpad + 2 * lg;

    v8f c = {};
    for (int k0 = 0; k0 < Kpad; k0 += 16) {       // Kpad % 16 == 0, no tail
        v2f a0 = *reinterpret_cast<const v2f*>(arow + k0);
        v2f b0 = *reinterpret_cast<const v2f*>(brow + k0);
        v2f a1 = *reinterpret_cast<const v2f*>(arow + k0 + 4);
        v2f b1 = *reinterpret_cast<const v2f*>(brow + k0 + 4);
        v2f a2 = *reinterpret_cast<const v2f*>(arow + k0 + 8);
        v2f b2 = *reinterpret_cast<const v2f*>(brow + k0 + 8);
        v2f a3 = *reinterpret_cast<const v2f*>(arow + k0 + 12);
        v2f b3 = *reinterpret_cast<const v2f*>(brow + k0 + 12);
        c = __builtin_amdgcn_wmma_f32_16x16x4_f32(false, a0, false, b0, (short)0, c, false, false);
        c = __builtin_amdgcn_wmma_f32_16x16x4_f32(false, a1, false, b1, (short)0, c, false, false);
        c = __builtin_amdgcn_wmma_f32_16x16x4_f32(false, a2, false, b2, (short)0, c, false, false);
        c = __builtin_amdgcn_wmma_f32_16x16x4_f32(false, a3, false, b3, (short)0, c, false, false);
    }

#pragma unroll
    for (int i = 0; i < 8; ++i) {
        const int row = mBase + i + 8 * lg;
        const float v = c[i] * dinv[row];         // fold dinv[src] scale
        hs [(size_t)row * outPad + n] = v;
        agg[(size_t)row * outPad + n] = v;        // self-loop init of agg
    }
}

// ---------------- edge scatter:  agg[dst] += hs[src]  ----------------------
__global__ void k_scatter_add(const int* __restrict__ src, const int* __restrict__ dst,
                              const float* __restrict__ hs, float* __restrict__ agg,
                              int out, int outPad) {
    const int total  = N_EDGES * out;             // <= 320M, fits int32
    const int stride = gridDim.x * blockDim.x;
    for (int t = blockIdx.x * blockDim.x + threadIdx.x; t < total; t += stride) {
        const int e = t / out;
        const int f = t - e * out;
        const float v = hs[(size_t)src[e] * outPad + f];
        atomicAdd(&agg[(size_t)dst[e] * outPad + f], v);
    }
}

// ---------------- finalize:  o = relu(agg*dinv + b) ------------------------
// dense=1 -> write compact [N,out] (final output); else in-place padded.
__global__ void k_finalize(const float* __restrict__ agg, const float* __restrict__ dinv,
                           const float* __restrict__ bias, float* __restrict__ o,
                           int out, int outPad, int do_relu, int dense) {
    const int total = N_NODES * out;
    const int t = blockIdx.x * blockDim.x + threadIdx.x;
    if (t >= total) return;
    const int row = t / out;
    const int f   = t - row * out;
    float v = agg[(size_t)row * outPad + f] * dinv[row] + bias[f];
    if (do_relu) v = fmaxf(v, 0.0f);
    if (dense) o[t] = v;
    else       o[(size_t)row * outPad + f] = v;
}

// ---------------------------------------------------------------------------
extern "C" void kernel_launch(void* const* d_in, const int* in_sizes, int n_in,
                              void* d_out, int out_size, void* d_ws, size_t ws_size,
                              hipStream_t stream) {
    (void)in_sizes; (void)n_in; (void)out_size; (void)ws_size;

    const float* x   = (const float*)d_in[0];
    const int*   ei  = (const int*)d_in[1];
    const int*   src = ei;                 // edge_index[0]
    const int*   dst = ei + N_EDGES;       // edge_index[1]
    const float* W[5]    = { (const float*)d_in[2], (const float*)d_in[4],
                             (const float*)d_in[6], (const float*)d_in[8],
                             (const float*)d_in[10] };
    const float* bias[5] = { (const float*)d_in[3], (const float*)d_in[5],
                             (const float*)d_in[7], (const float*)d_in[9],
                             (const float*)d_in[11] };

    static const int KIN   [5] = {512, 200, 175, 125, 75};
    static const int OUT   [5] = {200, 175, 125,  75, 50};
    static const int KPAD  [5] = {512, 208, 176, 128, 80};   // input strides
    static const int OUTPAD[5] = {208, 176, 128,  80, 64};

    // ws: dinv | B1(hs) | B2 | B3 | Wt0..Wt4
    float* dinv = (float*)d_ws;
    float* B1   = dinv + N_NODES;
    float* B2   = B1 + (size_t)N_NODES * MAX_OUTPAD;
    float* B3   = B2 + (size_t)N_NODES * MAX_OUTPAD;
    float* Wt[5];
    {
        float* p = B3 + (size_t)N_NODES * MAX_OUTPAD;
        for (int L = 0; L < 5; ++L) { Wt[L] = p; p += (size_t)KPAD[L] * OUTPAD[L]; }
    }

    // --- degree / dinv (shared by all layers) ---
    k_init_deg <<<(N_NODES + 255) / 256, 256, 0, stream>>>(dinv);
    k_accum_deg<<<(N_EDGES + 255) / 256, 256, 0, stream>>>(dst, dinv);
    k_rsqrt_deg<<<(N_NODES + 255) / 256, 256, 0, stream>>>(dinv);

    // --- pad + transpose weights once ---
    for (int L = 0; L < 5; ++L) {
        int total = KPAD[L] * OUTPAD[L];
        k_pad_weights<<<(total + 255) / 256, 256, 0, stream>>>(
            W[L], Wt[L], KIN[L], OUT[L], KPAD[L], OUTPAD[L]);
    }

    const float* cur = x;
    for (int L = 0; L < 5; ++L) {
        const int out = OUT[L], Kpad = KPAD[L], outPad = OUTPAD[L];
        const int nTiles = outPad / 16;
        float* hs  = B1;
        float* agg = (L & 1) ? B3 : B2;

        const int tiles = (N_NODES / 16) * nTiles;
        k_gemm_scale_wmma<<<(tiles + 3) / 4, 128, 0, stream>>>(
            cur, Wt[L], dinv, hs, agg, Kpad, outPad, nTiles);

        int swork = N_EDGES * out;
        int sblk  = (swork + 255) / 256;
        if (sblk > 262144) sblk = 262144;          // grid-stride covers rest
        k_scatter_add<<<sblk, 256, 0, stream>>>(src, dst, hs, agg, out, outPad);

        const int fwork = N_NODES * out;
        float* o = (L == 4) ? (float*)d_out : agg;
        k_finalize<<<(fwork + 255) / 256, 256, 0, stream>>>(
            agg, dinv, bias[L], o, out, outPad, (L < 4) ? 1 : 0, (L == 4) ? 1 : 0);
        cur = agg;
    }
}